// AttentionLayer_7533372637630
// MI455X (gfx1250) — compile-verified
//
#include <hip/hip_runtime.h>
#include <math.h>

// Problem dims (fixed by the reference)
#define BB  2048
#define NN  200
#define DD  128
#define HH  8
#define JD  1024        // D*H
#define RS  132         // padded LDS row stride (≡4 banks mod 64 -> conflict-free)

typedef float v2f __attribute__((ext_vector_type(2)));
typedef float v8f __attribute__((ext_vector_type(8)));

// ---------------------------------------------------------------------------
// Kernel 1: pq = tanh(q @ Wq.T + bq) -> workspace (B, 1024), j = d*H + h.
// Grid (B/16, 1024/128); 8 waves/block, one 16x16 WMMA tile per wave, K step 4.
// ---------------------------------------------------------------------------
__global__ __launch_bounds__(256) void pq_proj_kernel(
    const float* __restrict__ q,    // (B, 128)
    const float* __restrict__ Wq,   // (1024, 128)
    const float* __restrict__ bq,   // (1024)
    float* __restrict__ pq)         // (B, 1024)
{
    const int wave  = threadIdx.x >> 5;
    const int lane  = threadIdx.x & 31;
    const int lo    = lane & 15;
    const int hi    = lane >> 4;
    const int mbase = blockIdx.x * 16;
    const int jcol  = blockIdx.y * 128 + wave * 16 + lo;

    const float* arow = q  + (size_t)(mbase + lo) * DD + 2 * hi;
    const float* brow = Wq + (size_t)jcol        * DD + 2 * hi;   // B[k][j] = Wq[j][k]

    v8f c = {};
    #pragma unroll 4
    for (int kb = 0; kb < DD; kb += 4) {
        v2f a = *(const v2f*)(arow + kb);
        v2f b = *(const v2f*)(brow + kb);
        c = __builtin_amdgcn_wmma_f32_16x16x4_f32(
                false, a, false, b, (short)0, c, false, false);
    }

    const float bias = bq[jcol];
    #pragma unroll
    for (int i = 0; i < 8; ++i) {
        const int row = mbase + i + 8 * hi;
        pq[(size_t)row * JD + jcol] = tanhf(c[i] + bias);
    }
}

// ---------------------------------------------------------------------------
// Kernel 2: one block per batch b.
//  1) async-copy item[b] (200x128 f32) into LDS, coalesced, row stride 132;
//     rows 200..207 zero-filled.  GLOBAL_LOAD_ASYNC_TO_LDS_B128 + ASYNCcnt.
//  2) scores[n] = (item @ pq) @ Wr.T via WMMA f32 16x16x4 from LDS + shfl fold
//  3) reference-exact softmax (max over unmasked n<200, mask on exp, denom guard)
//  4) pooling out[d] = sum_n w[n]*item[n,d] from LDS.
// item is read from HBM exactly once, fully coalesced.
// ---------------------------------------------------------------------------
__global__ __launch_bounds__(256) void attn_pool_kernel(
    const float* __restrict__ item,          // (B, 200, 128)
    const unsigned char* __restrict__ mask,  // (B, 200, 1) bool
    const float* __restrict__ Wr,            // (1, 8)
    const float* __restrict__ pq,            // (B, 1024)
    float* __restrict__ out)                 // (B, 128)
{
    __shared__ __align__(16) float sh_item[208 * RS];  // ~107 KB
    __shared__ __align__(16) float pqp[2048];          // pair-interleaved pq (8 KB)
    __shared__ float scores[208];
    __shared__ float weightv[208];
    __shared__ float red[256];

    const int b    = blockIdx.x;
    const int tid  = threadIdx.x;
    const int wave = tid >> 5;
    const int lane = tid & 31;
    const int lo   = lane & 15;
    const int hi   = lane >> 4;

    const float* itemB = item + (size_t)b * (NN * DD);
    const float* pqB   = pq   + (size_t)b * JD;

    // ---- Stage item[b] -> LDS via async B128 copies (25 per thread) ----
    {
        const unsigned long long gbase = (unsigned long long)(const void*)itemB;
        #pragma unroll 5
        for (int i = tid; i < NN * (DD / 4); i += 256) {   // 6400 float4 tiles
            const int r = i >> 5;                          // row (128 floats = 32 f4)
            const int c = (i & 31) << 2;                   // float column
            const unsigned      lds = (unsigned)(uintptr_t)(&sh_item[r * RS + c]);
            const unsigned long long g = gbase + (unsigned long long)((r * DD + c) * 4);
            asm volatile("global_load_async_to_lds_b128 %0, %1, off"
                         :: "v"(lds), "v"(g) : "memory");
        }
        // zero-fill pad rows 200..207 (one float4 per thread)
        const int zr = NN + (tid >> 5);
        const int zc = (tid & 31) << 2;
        float4 z = {0.f, 0.f, 0.f, 0.f};
        *(float4*)&sh_item[zr * RS + zc] = z;
    }

    // ---- Stage pq[b] pair-interleaved: idx=(d/2)*32 + h*2 + (d&1), h>=8 -> 0 ----
    for (int i = tid; i < 2048; i += 256) {
        const int pr  = i >> 5;
        const int rem = i & 31;
        const int h   = rem >> 1;
        const int d   = pr * 2 + (rem & 1);
        pqp[i] = (h < HH) ? pqB[d * HH + h] : 0.0f;
    }

    asm volatile("s_wait_asynccnt 0x0" ::: "memory");
    __syncthreads();

    const float wrlane = (lo < HH) ? Wr[lo] : 0.0f;

    // ---- WMMA over 13 M-tiles (rows 200..207 are zeros) ----
    for (int tile = wave; tile < 13; tile += 8) {
        const int mbase = tile * 16;
        const float* arow = &sh_item[(mbase + lo) * RS + 2 * hi];

        v8f c = {};
        #pragma unroll 4
        for (int kb = 0; kb < DD; kb += 4) {
            v2f a  = *(const v2f*)(arow + kb);                       // ds_load_b64
            v2f bb = *(const v2f*)(&pqp[((kb >> 1) + hi) * 32 + lo * 2]);
            c = __builtin_amdgcn_wmma_f32_16x16x4_f32(
                    false, a, false, bb, (short)0, c, false, false);
        }
        #pragma unroll
        for (int i = 0; i < 8; ++i) {
            float r = c[i] * wrlane;
            r += __shfl_xor(r, 1);
            r += __shfl_xor(r, 2);
            r += __shfl_xor(r, 4);
            r += __shfl_xor(r, 8);          // groups of 16 lanes reduce separately
            const int n = mbase + i + 8 * hi;
            if (lo == 0) scores[n] = r;     // lane0 -> rows i, lane16 -> rows i+8
        }
    }
    __syncthreads();

    // ---- max over n<200 (unmasked, as in reference) ----
    red[tid] = (tid < NN) ? scores[tid] : -INFINITY;
    __syncthreads();
    for (int s = 128; s > 0; s >>= 1) {
        if (tid < s) red[tid] = fmaxf(red[tid], red[tid + s]);
        __syncthreads();
    }
    const float m = red[0];
    __syncthreads();

    // ---- att = exp(s - m)*mask ; denom guard ----
    float a = 0.0f;
    if (tid < NN) {
        const float mk = mask[(size_t)b * NN + tid] ? 1.0f : 0.0f;
        a = expf(scores[tid] - m) * mk;
    }
    red[tid] = a;
    __syncthreads();
    for (int s = 128; s > 0; s >>= 1) {
        if (tid < s) red[tid] += red[tid + s];
        __syncthreads();
    }
    const float denom = red[0];
    const float g = (denom < 1e-7f) ? denom + 1.0f : denom;
    if (tid < NN) weightv[tid] = a / g;
    __syncthreads();

    // ---- pooling from LDS: threads 0..127 n=0..99, 128..255 n=100..199 ----
    const int d  = tid & 127;
    const int g2 = tid >> 7;
    float acc = 0.0f;
    #pragma unroll 4
    for (int n = g2 * 100; n < g2 * 100 + 100; ++n)
        acc += weightv[n] * sh_item[n * RS + d];
    red[tid] = acc;
    __syncthreads();
    if (tid < 128)
        out[(size_t)b * DD + tid] = red[tid] + red[tid + 128];
}

// ---------------------------------------------------------------------------
extern "C" void kernel_launch(void* const* d_in, const int* in_sizes, int n_in,
                              void* d_out, int out_size, void* d_ws, size_t ws_size,
                              hipStream_t stream) {
    const float*         item = (const float*)d_in[0];         // (2048,200,128)
    const float*         q    = (const float*)d_in[1];         // (2048,128)
    const unsigned char* mask = (const unsigned char*)d_in[2]; // (2048,200,1) bool
    const float*         Wq   = (const float*)d_in[3];         // (1024,128)
    const float*         bq   = (const float*)d_in[4];         // (1024)
    const float*         Wr   = (const float*)d_in[5];         // (1,8)
    float*               out  = (float*)d_out;                 // (2048,128)

    float* pq = (float*)d_ws;   // needs 2048*1024*4 = 8 MB workspace

    dim3 grid1(BB / 16, JD / 128);
    pq_proj_kernel<<<grid1, 256, 0, stream>>>(q, Wq, bq, pq);

    attn_pool_kernel<<<BB, 256, 0, stream>>>(item, mask, Wr, pq, out);
}